// BatchNeighborList_42898133353121
// MI455X (gfx1250) — compile-verified
//
#include <hip/hip_runtime.h>
#include <hip/hip_bf16.h>
#include <math.h>

typedef __attribute__((ext_vector_type(2))) float v2f;
typedef __attribute__((ext_vector_type(3))) float v3f;
typedef __attribute__((ext_vector_type(8))) float v8f;

#define CUTOFF   5.0f
#define WRAP_EPS 1e-7f
#define MIN_DIST 0.01f

// ---------------------------------------------------------------------------
// Setup: per system, compute inv(cell), wrapped positions wp, pack wp4 =
// (x, y, z, |wp|^2), and the 27 cartesian shift vectors sc = shifts @ cell.
// ---------------------------------------------------------------------------
__global__ void __launch_bounds__(256)
nbl_setup_kernel(const float* __restrict__ pos,   // [B*N,3]
                 const float* __restrict__ cell,  // [B,3,3] rows = lattice vecs
                 float* __restrict__ wp4,         // [B,N,4]
                 float* __restrict__ sc,          // [B,27,3]
                 int N) {
    const int b = blockIdx.x;
    const float* C = cell + b * 9;
    const float m00 = C[0], m01 = C[1], m02 = C[2];
    const float m10 = C[3], m11 = C[4], m12 = C[5];
    const float m20 = C[6], m21 = C[7], m22 = C[8];
    const float det = m00 * (m11 * m22 - m12 * m21)
                    - m01 * (m10 * m22 - m12 * m20)
                    + m02 * (m10 * m21 - m11 * m20);
    const float rd = 1.0f / det;
    const float i00 = (m11 * m22 - m12 * m21) * rd;
    const float i01 = (m02 * m21 - m01 * m22) * rd;
    const float i02 = (m01 * m12 - m02 * m11) * rd;
    const float i10 = (m12 * m20 - m10 * m22) * rd;
    const float i11 = (m00 * m22 - m02 * m20) * rd;
    const float i12 = (m02 * m10 - m00 * m12) * rd;
    const float i20 = (m10 * m21 - m11 * m20) * rd;
    const float i21 = (m01 * m20 - m00 * m21) * rd;
    const float i22 = (m00 * m11 - m01 * m10) * rd;

    for (int i = threadIdx.x; i < N; i += blockDim.x) {
        const float px = pos[(size_t)(b * N + i) * 3 + 0];
        const float py = pos[(size_t)(b * N + i) * 3 + 1];
        const float pz = pos[(size_t)(b * N + i) * 3 + 2];
        float s0 = px * i00 + py * i10 + pz * i20 + WRAP_EPS;
        float s1 = px * i01 + py * i11 + pz * i21 + WRAP_EPS;
        float s2 = px * i02 + py * i12 + pz * i22 + WRAP_EPS;
        s0 = s0 - floorf(s0) - WRAP_EPS;
        s1 = s1 - floorf(s1) - WRAP_EPS;
        s2 = s2 - floorf(s2) - WRAP_EPS;
        const float wx = s0 * m00 + s1 * m10 + s2 * m20;
        const float wy = s0 * m01 + s1 * m11 + s2 * m21;
        const float wz = s0 * m02 + s1 * m12 + s2 * m22;
        float* o = wp4 + (size_t)(b * N + i) * 4;
        o[0] = wx; o[1] = wy; o[2] = wz;
        o[3] = wx * wx + wy * wy + wz * wz;
    }
    for (int s = threadIdx.x; s < 27; s += blockDim.x) {
        const float a0 = (float)(s / 9 - 1);
        const float a1 = (float)((s / 3) % 3 - 1);
        const float a2 = (float)(s % 3 - 1);
        float* o = sc + (size_t)(b * 27 + s) * 3;
        o[0] = a0 * m00 + a1 * m10 + a2 * m20;
        o[1] = a0 * m01 + a1 * m11 + a2 * m21;
        o[2] = a0 * m02 + a1 * m12 + a2 * m22;
    }
}

// ---------------------------------------------------------------------------
// Main kernel: one wave computes a 16(i) x 16(c=(j,s)) tile.
//   dist^2(i,c) = |wp_i|^2 - 2 wp_i . q_c + |q_c|^2,  q_c = wp_j + shift_s
// Bilinear part via one V_WMMA_F32_16X16X4_F32.  Mask computed on dist^2
// (no dependency on the transcendental), applied as a multiply.  All output
// through saddr-form non-temporal stores with 32-bit strided offsets.
// ---------------------------------------------------------------------------
__global__ void __launch_bounds__(256)
nbl_tile_kernel(const float* __restrict__ wp4,   // [B,N,4]
                const float* __restrict__ sc,    // [B,27,3]
                float* __restrict__ diff_out,    // [B,N,NC,3]
                float* __restrict__ dist_out,    // [B,N,NC]
                float* __restrict__ mask_out,    // [B,N,NC]
                int N) {
    const int NC   = N * 27;
    const int b    = blockIdx.z;
    const int i0   = blockIdx.y * 16;
    const int wave = threadIdx.x >> 5;
    const int lane = threadIdx.x & 31;
    const int r    = lane & 15;
    const int hi   = lane >> 4;

    // Stage the 16 wp4 rows of this i-tile (shared by all 8 waves).
    __shared__ float4 swp[16];
    if (threadIdx.x < 64)
        ((float*)swp)[threadIdx.x] = wp4[(size_t)(b * N + i0) * 4 + threadIdx.x];
    __syncthreads();

    const int c0 = (blockIdx.x * 8 + wave) * 16;
    if (c0 >= NC) return;                      // wave-uniform: EXEC stays full
    const int c = c0 + r;
    const int j = c / 27;
    const int s = c - j * 27;

    // Column vector q_c = wp_j + shift_s and its norm.
    const float* wj = wp4 + (size_t)(b * N + j) * 4;
    const float* ss = sc + (size_t)(b * 27 + s) * 3;
    const float qx = wj[0] + ss[0];
    const float qy = wj[1] + ss[1];
    const float qz = wj[2] + ss[2];
    const float qn = qx * qx + qy * qy + qz * qz;

    // A operand: lane holds (K=2*hi, K=2*hi+1) of packed row (x,y,z,|wp|^2).
    const float* wi = wp4 + (size_t)(b * N + i0 + r) * 4 + 2 * hi;
    v2f av; av.x = wi[0]; av.y = wi[1];
    // B operand: lane holds (K=2*hi, K=2*hi+1) of column c.
    v2f bv;
    bv.x = (hi == 0) ? (-2.0f * qx) : (-2.0f * qz);
    bv.y = (hi == 0) ? (-2.0f * qy) : 1.0f;

    v8f acc = {};
    acc = __builtin_amdgcn_wmma_f32_16x16x4_f32(false, av, false, bv,
                                                (short)0, acc, false, false);

    // D lane layout: column r, rows m = v + 8*hi for v = 0..7.
    // 32-bit strided offsets (sections < 2^31 bytes) -> saddr-form stores.
    const unsigned NCu   = (unsigned)NC;
    unsigned idx   = (unsigned)((b * N + i0 + 8 * hi)) * NCu + (unsigned)c;
    unsigned dbyte = idx * 12u;                 // diff byte offset (idx*3 floats)
    const unsigned dstep = NCu * 12u;

#pragma unroll
    for (int v = 0; v < 8; ++v) {
        const float4 w = swp[v + 8 * hi];       // unconditional ds_load_b96
        const float d2 = acc[v] + qn;
        // mask on dist^2: dist<5 && dist>0.01  <=>  d2<25 && d2>1e-4
        const bool  mk  = (d2 < CUTOFF * CUTOFF) && (d2 > MIN_DIST * MIN_DIST);
        const float mmv = mk ? 1.0f : 0.0f;     // mask as multiplier: no branches
        const float dist = __builtin_amdgcn_sqrtf(fmaxf(d2, 0.0f)) * mmv;
        v3f dv;
        dv.x = (qx - w.x) * mmv;
        dv.y = (qy - w.y) * mmv;
        dv.z = (qz - w.z) * mmv;
        __builtin_nontemporal_store(dist, dist_out + idx);
        __builtin_nontemporal_store(mmv,  mask_out + idx);
        // Contiguous 12B/lane NT store: saddr base + 32-bit voffset.
        asm volatile("global_store_b96 %0, %1, %2 th:TH_STORE_NT"
                     :: "v"(dbyte), "v"(dv), "s"(diff_out) : "memory");
        idx   += NCu;
        dbyte += dstep;
    }
}

extern "C" void kernel_launch(void* const* d_in, const int* in_sizes, int n_in,
                              void* d_out, int out_size, void* d_ws, size_t ws_size,
                              hipStream_t stream) {
    const float* positions = (const float*)d_in[0];   // [B*N,3]
    const float* cell      = (const float*)d_in[1];   // [B,3,3]
    // d_in[2] = n_atoms (all equal N) — unused.

    const int B = in_sizes[1] / 9;
    const int N = in_sizes[0] / (3 * B);
    const int NC = N * 27;

    float* wp4 = (float*)d_ws;
    float* sc  = wp4 + (size_t)B * N * 4;

    // Output sections: diff [B,N,NC,3] | dist [B,N,NC] | mask [B,N,NC]
    float* out      = (float*)d_out;
    const size_t nPair = (size_t)B * N * (size_t)NC;
    float* diff_out = out;
    float* dist_out = out + nPair * 3;
    float* mask_out = out + nPair * 4;

    nbl_setup_kernel<<<B, 256, 0, stream>>>(positions, cell, wp4, sc, N);

    const int iTiles = (N + 15) / 16;
    const int cTiles = (NC + 15) / 16;
    dim3 grid((cTiles + 7) / 8, iTiles, B);
    nbl_tile_kernel<<<grid, 256, 0, stream>>>(wp4, sc, diff_out, dist_out,
                                              mask_out, N);
}